// get_loss_23957327577678
// MI455X (gfx1250) — compile-verified
//
#include <hip/hip_runtime.h>
#include <hip/hip_bf16.h>

typedef __attribute__((ext_vector_type(2))) float v2f;
typedef __attribute__((ext_vector_type(8))) float v8f;

#define WAVES_PER_BLOCK 8
#define BLOCK_THREADS   (WAVES_PER_BLOCK * 32)

// One wave owns a 16-row tile of P (queries) and sweeps all 16-col tiles of Q
// (targets), computing   min_j max(||p_i - q_j||^2, 0)   for its 16 rows via
// V_WMMA_F32_16X16X4_F32 (K=3 padded to 4), then writes one partial sum.
__global__ __launch_bounds__(BLOCK_THREADS)
void chamfer_rowmin_kernel(const float* __restrict__ P,  // [B, NP, 3] query rows
                           const float* __restrict__ Q,  // [B, NQ, 3] targets
                           float* __restrict__ partials, // [B * NP/16]
                           int NP, int NQ, int tilesPerBatch /* = NP/16 */,
                           int totalWaves) {
    const int lane = threadIdx.x & 31;
    const int wave = blockIdx.x * WAVES_PER_BLOCK + (threadIdx.x >> 5);
    if (wave >= totalWaves) return;   // uniform per-wave branch; EXEC stays all-1s

    const int b      = wave / tilesPerBatch;
    const int ntile  = wave - b * tilesPerBatch;
    const int subl   = lane & 15;         // 0..15: row (A) / column (B,C) index
    const bool hi    = lane >= 16;        // high half-wave holds K=2,3 for A/B

    // ---- Load this wave's 16-row A tile of P, build A operand (16x4 f32) ----
    const int row = (ntile << 4) + subl;
    const float* xp = P + ((size_t)b * NP + row) * 3;
    const float x0 = xp[0], x1 = xp[1], x2e = xp[2];
    const float xsq = fmaf(x0, x0, fmaf(x1, x1, x2e * x2e));

    v2f amat;
    amat.x = hi ? x2e : x0;   // VGPR0: K=0 (lanes 0-15) / K=2 (lanes 16-31)
    amat.y = hi ? 0.f : x1;   // VGPR1: K=1 (lanes 0-15) / K=3 = pad 0

    // x2 aligned to C/D layout: C[r] of this lane corresponds to row r (low
    // half) or row r+8 (high half). xsq of row q lives in lane q.
    float x2c[8];
#pragma unroll
    for (int r = 0; r < 8; ++r)
        x2c[r] = __shfl(xsq, hi ? (r + 8) : r, 32);

    float acc[8];
#pragma unroll
    for (int r = 0; r < 8; ++r) acc[r] = 3.0e38f;

    const int mtiles = NQ >> 4;
    const float* qbase = Q + (size_t)b * NQ * 3;

    for (int mt = 0; mt < mtiles; ++mt) {
        // Each lane loads target point for its column (same point both halves)
        const float* yp = qbase + (size_t)((mt << 4) + subl) * 3;
        const float y0 = yp[0], y1 = yp[1], y2e = yp[2];
        const float ysq = fmaf(y0, y0, fmaf(y1, y1, y2e * y2e));

        // B operand = -2 * Q^T  (4x16 f32): VGPR0 K=0/K=2, VGPR1 K=1/K=3
        v2f bmat;
        bmat.x = -2.0f * (hi ? y2e : y0);
        bmat.y = hi ? 0.0f : (-2.0f * y1);

        // C = ||y_j||^2 broadcast down every row of the tile
        v8f c = { ysq, ysq, ysq, ysq, ysq, ysq, ysq, ysq };

        // D[i][j] = ||y_j||^2 - 2 * x_i . y_j
        v8f d = __builtin_amdgcn_wmma_f32_16x16x4_f32(
            /*neg_a=*/false, amat, /*neg_b=*/false, bmat,
            /*c_mod=*/(short)0, c, /*reuse_a=*/false, /*reuse_b=*/false);

#pragma unroll
        for (int r = 0; r < 8; ++r) acc[r] = fminf(acc[r], d[r]);
    }

    // ---- Min across the 16 column-lanes (per half-wave butterfly) ----
#pragma unroll
    for (int off = 1; off < 16; off <<= 1) {
#pragma unroll
        for (int r = 0; r < 8; ++r)
            acc[r] = fminf(acc[r], __shfl_xor(acc[r], off, 32));
    }

    // Lanes 0 and 16 now hold row-mins for rows 0-7 / 8-15 of the tile.
    float psum = 0.0f;
#pragma unroll
    for (int r = 0; r < 8; ++r)
        psum += fmaxf(acc[r] + x2c[r], 0.0f);    // add ||x||^2, clamp, sum

    const float other = __shfl(psum, 16, 32);    // high-half partial -> lane 0
    if (lane == 0) partials[wave] = psum + other;
}

// Deterministic fixed-order reduction of both partial arrays + scalar epilogue.
__global__ __launch_bounds__(256)
void chamfer_finalize_kernel(const float* __restrict__ px, int nx,
                             const float* __restrict__ py, int ny,
                             const float* __restrict__ bpp,
                             const float* __restrict__ lam,
                             float* __restrict__ out,
                             float inv_x, float inv_y) {
    __shared__ float sx[256];
    __shared__ float sy[256];
    const int t = threadIdx.x;
    float ax = 0.0f, ay = 0.0f;
    for (int i = t; i < nx; i += 256) ax += px[i];
    for (int i = t; i < ny; i += 256) ay += py[i];
    sx[t] = ax; sy[t] = ay;
    __syncthreads();
    for (int s = 128; s > 0; s >>= 1) {
        if (t < s) { sx[t] += sx[t + s]; sy[t] += sy[t + s]; }
        __syncthreads();
    }
    if (t == 0)
        out[0] = sx[0] * inv_x + sy[0] * inv_y + lam[0] * bpp[0];
}

extern "C" void kernel_launch(void* const* d_in, const int* in_sizes, int n_in,
                              void* d_out, int out_size, void* d_ws, size_t ws_size,
                              hipStream_t stream) {
    const float* pc_pred   = (const float*)d_in[0];   // [B, N, 3]
    const float* pc_target = (const float*)d_in[1];   // [B, M, 3]
    const float* bpp       = (const float*)d_in[2];   // scalar
    const float* lam       = (const float*)d_in[3];   // scalar
    float* out = (float*)d_out;

    const int B = 4;                                  // fixed by reference
    const int N = in_sizes[0] / (B * 3);
    const int M = in_sizes[1] / (B * 3);

    const int tilesX = N >> 4;                        // row tiles of pred
    const int tilesY = M >> 4;                        // row tiles of target
    const int wavesX = B * tilesX;
    const int wavesY = B * tilesY;

    float* ws = (float*)d_ws;
    float* partX = ws;                                // [wavesX]
    float* partY = ws + wavesX;                       // [wavesY]

    const int blocksX = (wavesX + WAVES_PER_BLOCK - 1) / WAVES_PER_BLOCK;
    const int blocksY = (wavesY + WAVES_PER_BLOCK - 1) / WAVES_PER_BLOCK;

    // Pass 1: x -> nearest y
    chamfer_rowmin_kernel<<<blocksX, BLOCK_THREADS, 0, stream>>>(
        pc_pred, pc_target, partX, N, M, tilesX, wavesX);
    // Pass 2: y -> nearest x (symmetric swap)
    chamfer_rowmin_kernel<<<blocksY, BLOCK_THREADS, 0, stream>>>(
        pc_target, pc_pred, partY, M, N, tilesY, wavesY);

    chamfer_finalize_kernel<<<1, 256, 0, stream>>>(
        partX, wavesX, partY, wavesY, bpp, lam, out,
        1.0f / (float)(B * N), 1.0f / (float)(B * M));
}